// DenseEquivariantDecoder_1683627180252
// MI455X (gfx1250) — compile-verified
//
#include <hip/hip_runtime.h>
#include <math.h>
#include <stdint.h>

// Problem constants (from reference)
#define BB 8
#define NN 96
#define FF 128
#define BN (BB * NN)     // 768 nodes
#define F4 (4 * FF)      // 512
#define F2 (2 * FF)      // 256
#define F3 (3 * FF)      // 384
#define NRBF 20
#define NPAIR (BB * NN * NN)  // 73728 ordered (b,i,j) pairs incl diagonal
#define PI_F 3.14159265358979323846f

typedef __attribute__((ext_vector_type(2))) float v2f;
typedef __attribute__((ext_vector_type(8))) float v8f;

#if defined(__HIP_DEVICE_COMPILE__) && __has_builtin(__builtin_amdgcn_tensor_load_to_lds)
#define USE_TDM 1
#else
#define USE_TDM 0
#endif

__device__ __forceinline__ float swish_f(float x) {
  return x / (1.0f + __expf(-x));
}

// ---------------------------------------------------------------------------
// f32 WMMA GEMM: C[M x Nc] = act(A[M x K] @ W[K x Nc] + bias[Nc])
// One wave owns a 16x32 tile (two 16x16 accumulators sharing one A fragment),
// K stepped by 4 through V_WMMA_F32_16X16X4_F32.
// M mult of 16, Nc mult of 32, K mult of 4.
// Layouts per CDNA5 ISA 7.12.2:
//   A 16x4 : lane<16 -> K={k,k+1}, lane>=16 -> K={k+2,k+3}; M = lane&15
//   B 4x16 : mirrored (lane-half selects K pair, VGPR selects within pair)
//   C 16x16: vgpr r -> M = r + (lane>>4)*8 ; N = lane&15
// ---------------------------------------------------------------------------
__global__ void gemm_wmma(const float* __restrict__ A, const float* __restrict__ W,
                          const float* __restrict__ bias, float* __restrict__ C,
                          int M, int K, int Nc, int tiles2, int act) {
  int wave = (int)((blockIdx.x * blockDim.x + threadIdx.x) >> 5);
  if (wave >= tiles2) return;               // wave-uniform: EXEC stays all-1s
  int lane = (int)(threadIdx.x & 31);
  int ntN2 = Nc >> 5;                       // # of 32-wide column tiles
  int tm = wave / ntN2;
  int tn2 = wave - tm * ntN2;
  int half = lane >> 4;
  int l15 = lane & 15;

  const float* arow = A + (size_t)(tm * 16 + l15) * (size_t)K;
  int n0 = tn2 * 32 + l15;

  v8f acc0 = {};
  v8f acc1 = {};
  for (int k = 0; k < K; k += 4) {
    int ka = k + half * 2;
    v2f av, b0, b1;
    av.x = arow[ka];
    av.y = arow[ka + 1];
    const float* wr0 = W + (size_t)ka * Nc;
    const float* wr1 = W + (size_t)(ka + 1) * Nc;
    b0.x = wr0[n0];
    b0.y = wr1[n0];
    b1.x = wr0[n0 + 16];
    b1.y = wr1[n0 + 16];
    acc0 = __builtin_amdgcn_wmma_f32_16x16x4_f32(false, av, false, b0, (short)0, acc0, false, false);
    acc1 = __builtin_amdgcn_wmma_f32_16x16x4_f32(false, av, false, b1, (short)0, acc1, false, false);
  }
  float bv0 = bias ? bias[n0] : 0.0f;
  float bv1 = bias ? bias[n0 + 16] : 0.0f;
#pragma unroll
  for (int r = 0; r < 8; ++r) {
    int m = tm * 16 + r + half * 8;
    float v0 = acc0[r] + bv0;
    float v1 = acc1[r] + bv1;
    if (act) { v0 = swish_f(v0); v1 = swish_f(v1); }
    C[(size_t)m * Nc + n0] = v0;
    C[(size_t)m * Nc + n0 + 16] = v1;
  }
}

// ---------------------------------------------------------------------------
// Precompute: inverse sqrt degree per node
// ---------------------------------------------------------------------------
__global__ void deg_kernel(const float* __restrict__ adj, float* __restrict__ dinv) {
  int n = blockIdx.x * blockDim.x + threadIdx.x;
  if (n >= BN) return;
  const float* row = adj + (size_t)n * NN;
  float s = 0.0f;
  for (int j = 0; j < NN; ++j) s += row[j];
  dinv[n] = rsqrtf(s);
}

// ---------------------------------------------------------------------------
// Precompute per ordered pair (b,i,j): unit vector i->j, RBF*env*wgt, env*wgt.
// Diagonal entries written as zeros -> diagonal contributes exactly zero in
// the message loop, so it needs no branch there.
// ---------------------------------------------------------------------------
__global__ void edge_precompute(const float* __restrict__ xyz, const float* __restrict__ dinv,
                                float* __restrict__ unitv, float* __restrict__ rbfe,
                                float* __restrict__ ew) {
  int p = blockIdx.x * blockDim.x + threadIdx.x;
  if (p >= NPAIR) return;
  int j = p % NN;
  int bi = p / NN;
  int i = bi % NN;
  int b = bi / NN;
  if (i == j) {
    unitv[p * 3 + 0] = 0.f; unitv[p * 3 + 1] = 0.f; unitv[p * 3 + 2] = 0.f;
    ew[p] = 0.f;
    for (int r = 0; r < NRBF; ++r) rbfe[p * NRBF + r] = 0.f;
    return;
  }
  int ni = b * NN + i, nj = b * NN + j;
  float dx = xyz[nj * 3 + 0] - xyz[ni * 3 + 0];
  float dy = xyz[nj * 3 + 1] - xyz[ni * 3 + 1];
  float dz = xyz[nj * 3 + 2] - xyz[ni * 3 + 2];
  float dist = sqrtf(dx * dx + dy * dy + dz * dz);
  float inv = 1.0f / dist;
  unitv[p * 3 + 0] = dx * inv;
  unitv[p * 3 + 1] = dy * inv;
  unitv[p * 3 + 2] = dz * inv;
  float wgt = dinv[ni] * dinv[nj];
  float env = (dist <= 10.0f) ? 0.5f * (cosf(PI_F * dist * 0.1f) + 1.0f) : 0.0f;
  float s = env * wgt;
  ew[p] = s;
  float c = inv * s;
  for (int r = 0; r < NRBF; ++r)
    rbfe[p * NRBF + r] = sinf((float)(r + 1) * PI_F * dist * 0.1f) * c;
}

// ---------------------------------------------------------------------------
// Message pass: one 128-thread block per node (b,i); accumulates its 95 edges
// in registers (no atomics). msg_wd (40KB) staged into LDS via the Tensor
// Data Mover (wave 0 issues tensor_load_to_lds, waits TENSORcnt); per-pair
// scalars staged by plain cooperative loads. Edge loop unrolled x4 so each
// s_wd LDS read feeds 4 edges. V layout [node][d][f]; Hs in place; V
// double-buffered (v_j is read across blocks).
// ---------------------------------------------------------------------------
__global__ void message_kernel(const float* __restrict__ phi,      // [768][512]
                               float* __restrict__ Hs,             // [768][128] +=
                               const float* __restrict__ V_in,     // [768][3][128]
                               float* __restrict__ V_out,          // [768][3][128]
                               const float* __restrict__ unitv,    // [NPAIR][3]
                               const float* __restrict__ rbfe,     // [NPAIR][20]
                               const float* __restrict__ ew,       // [NPAIR]
                               const float* __restrict__ msg_wd,   // [20][512]
                               const float* __restrict__ msg_bd) { // [512]
  __shared__ float s_wd[NRBF * F4];   // 40 KB
  __shared__ float s_bd[F4];
  __shared__ float s_unit[NN * 3];
  __shared__ float s_rbf[NN * NRBF];
  __shared__ float s_ew[NN];

  int node = blockIdx.x;              // 0..767
  int b = node / NN;
  int i = node - b * NN;
  (void)i;
  int t = threadIdx.x;                // 0..127 (feature index)
  int pbase = node * NN;

#if USE_TDM
  if (threadIdx.x < 32) {             // wave 0 issues the DMA (EXEC-independent)
    typedef __attribute__((ext_vector_type(4))) unsigned int u32x4;
    typedef __attribute__((ext_vector_type(8))) int i32x8;
    typedef __attribute__((ext_vector_type(4))) int i32x4;
    const unsigned int NELEM = (unsigned int)(NRBF * F4);  // 10240 f32
    unsigned long long ga = (unsigned long long)(uintptr_t)msg_wd;
    unsigned int lds_off = (unsigned int)(uintptr_t)(&s_wd[0]);  // LDS aperture low bits
    u32x4 g0;
    g0[0] = 1u;                                            // count=1 (valid), user mode
    g0[1] = lds_off;                                       // lds_addr
    g0[2] = (unsigned int)(ga & 0xFFFFFFFFu);              // global_addr[31:0] (bit 64+)
    g0[3] = (unsigned int)((ga >> 32) & 0x01FFFFFFu) | 0x80000000u;  // addr[56:32] | type=2
    i32x8 g1;
    g1[0] = (int)(2u << 16);                // wg_mask=0 | data_size=2 (4B)
    g1[1] = (int)((NELEM & 0xFFFFu) << 16); // tensor_dim0 low16 @ bits[63:48]
    g1[2] = (int)(((NELEM >> 16) & 0xFFFFu)) | (1 << 16);  // dim0 hi | tensor_dim1=1
    g1[3] = (int)((NELEM & 0xFFFFu) << 16); // dim1 hi(0) | tile_dim0 @ bits[127:112]
    g1[4] = 1;                              // tile_dim1=1, tile_dim2=0
    g1[5] = (int)NELEM;                     // tensor_dim0_stride low32
    g1[6] = 0;                              // stride hi | tensor_dim1_stride lo
    g1[7] = 0;
    i32x4 g2 = {};                          // unused (2D tensor)
    i32x4 g3 = {};
    i32x8 g4 = {};                          // 6-arg clang-23 form: extra group
    __builtin_amdgcn_tensor_load_to_lds(g0, g1, g2, g3, g4, 0);
    __builtin_amdgcn_s_wait_tensorcnt(0);
  }
#else
  for (int idx = t; idx < NRBF * F4; idx += FF) s_wd[idx] = msg_wd[idx];
#endif
  for (int idx = t; idx < F4; idx += FF) s_bd[idx] = msg_bd[idx];
  for (int idx = t; idx < NN * 3; idx += FF) s_unit[idx] = unitv[(size_t)pbase * 3 + idx];
  for (int idx = t; idx < NN * NRBF; idx += FF) s_rbf[idx] = rbfe[(size_t)pbase * NRBF + idx];
  if (t < NN) s_ew[t] = ew[pbase + t];
  __syncthreads();

  float vi0 = V_in[((size_t)node * 3 + 0) * FF + t];
  float vi1 = V_in[((size_t)node * 3 + 1) * FF + t];
  float vi2 = V_in[((size_t)node * 3 + 2) * FF + t];
  float bd0 = s_bd[0 * FF + t];
  float bd1 = s_bd[1 * FF + t];
  float bd2 = s_bd[2 * FF + t];
  float bd3 = s_bd[3 * FF + t];

  float accS = 0.0f, accV0 = 0.0f, accV1 = 0.0f, accV2 = 0.0f;

  for (int j0 = 0; j0 < NN; j0 += 4) {
    // prefetch next group's phi rows
    __builtin_prefetch((const void*)(phi + ((size_t)(b * NN + ((j0 + 4) % NN))) * F4 + t), 0, 1);

    float ws0[4], ws1[4], ws2[4], ws3[4];
#pragma unroll
    for (int u = 0; u < 4; ++u) {
      float ewj = s_ew[j0 + u];
      ws0[u] = bd0 * ewj;
      ws1[u] = bd1 * ewj;
      ws2[u] = bd2 * ewj;
      ws3[u] = bd3 * ewj;
    }
#pragma unroll
    for (int r = 0; r < NRBF; ++r) {
      float w0 = s_wd[r * F4 + 0 * FF + t];
      float w1 = s_wd[r * F4 + 1 * FF + t];
      float w2 = s_wd[r * F4 + 2 * FF + t];
      float w3 = s_wd[r * F4 + 3 * FF + t];
#pragma unroll
      for (int u = 0; u < 4; ++u) {
        float rv = s_rbf[(j0 + u) * NRBF + r];
        ws0[u] = fmaf(rv, w0, ws0[u]);
        ws1[u] = fmaf(rv, w1, ws1[u]);
        ws2[u] = fmaf(rv, w2, ws2[u]);
        ws3[u] = fmaf(rv, w3, ws3[u]);
      }
    }
#pragma unroll
    for (int u = 0; u < 4; ++u) {
      int dst = b * NN + j0 + u;
      const float* prow = phi + (size_t)dst * F4;
      float a0 = prow[0 * FF + t] * ws0[u];
      float a1 = prow[1 * FF + t] * ws1[u];
      float a2 = prow[2 * FF + t] * ws2[u];
      float a3 = prow[3 * FF + t] * ws3[u];

      float vj0 = V_in[((size_t)dst * 3 + 0) * FF + t];
      float vj1 = V_in[((size_t)dst * 3 + 1) * FF + t];
      float vj2 = V_in[((size_t)dst * 3 + 2) * FF + t];

      float c0 = vi1 * vj2 - vi2 * vj1;
      float c1 = vi2 * vj0 - vi0 * vj2;
      float c2 = vi0 * vj1 - vi1 * vj0;

      float u0 = s_unit[(j0 + u) * 3 + 0];
      float u1 = s_unit[(j0 + u) * 3 + 1];
      float u2 = s_unit[(j0 + u) * 3 + 2];

      accV0 += a2 * u0 + a0 * vj0 + a3 * c0;
      accV1 += a2 * u1 + a0 * vj1 + a3 * c1;
      accV2 += a2 * u2 + a0 * vj2 + a3 * c2;
      accS  += a1;
    }
  }

  Hs[(size_t)node * FF + t] += accS;
  V_out[((size_t)node * 3 + 0) * FF + t] = vi0 + accV0;
  V_out[((size_t)node * 3 + 1) * FF + t] = vi1 + accV1;
  V_out[((size_t)node * 3 + 2) * FF + t] = vi2 + accV2;
}

// ---------------------------------------------------------------------------
__global__ void vnorm_stack(const float* __restrict__ Hs, const float* __restrict__ vv,
                            float* __restrict__ sstack) {
  int n = blockIdx.x;
  int t = threadIdx.x;
  float x0 = vv[((size_t)n * 3 + 0) * FF + t];
  float x1 = vv[((size_t)n * 3 + 1) * FF + t];
  float x2 = vv[((size_t)n * 3 + 2) * FF + t];
  float nr = sqrtf(x0 * x0 + x1 * x1 + x2 * x2 + 3e-15f);
  sstack[(size_t)n * F2 + t] = Hs[(size_t)n * FF + t];
  sstack[(size_t)n * F2 + FF + t] = nr;
}

__global__ void final_update(float* __restrict__ Hs, float* __restrict__ V,
                             const float* __restrict__ uv, const float* __restrict__ vv,
                             const float* __restrict__ amat) {
  int n = blockIdx.x;
  int t = threadIdx.x;
  float u0 = uv[((size_t)n * 3 + 0) * FF + t];
  float u1 = uv[((size_t)n * 3 + 1) * FF + t];
  float u2 = uv[((size_t)n * 3 + 2) * FF + t];
  float w0 = vv[((size_t)n * 3 + 0) * FF + t];
  float w1 = vv[((size_t)n * 3 + 1) * FF + t];
  float w2 = vv[((size_t)n * 3 + 2) * FF + t];
  float dot = u0 * w0 + u1 * w1 + u2 * w2;
  float ga = amat[(size_t)n * F3 + 0 * FF + t];
  float gb = amat[(size_t)n * F3 + 1 * FF + t];
  float gc = amat[(size_t)n * F3 + 2 * FF + t];
  Hs[(size_t)n * FF + t] += gb * dot + gc;
  V[((size_t)n * 3 + 0) * FF + t] += u0 * ga;
  V[((size_t)n * 3 + 1) * FF + t] += u1 * ga;
  V[((size_t)n * 3 + 2) * FF + t] += u2 * ga;
}

__global__ void copy_f32(const float* __restrict__ src, float* __restrict__ dst, int n) {
  int idx = blockIdx.x * blockDim.x + threadIdx.x;
  if (idx < n) dst[idx] = src[idx];
}

__global__ void zero_f32(float* __restrict__ dst, int n) {
  int idx = blockIdx.x * blockDim.x + threadIdx.x;
  if (idx < n) dst[idx] = 0.0f;
}

// out = [Hs (B,N,F)] ++ [V transposed to (B,N,F,3)]
__global__ void write_out(const float* __restrict__ Hs, const float* __restrict__ V,
                          float* __restrict__ out) {
  int n = blockIdx.x;
  int t = threadIdx.x;
  out[(size_t)n * FF + t] = Hs[(size_t)n * FF + t];
  float* oV = out + (size_t)BN * FF;
#pragma unroll
  for (int d = 0; d < 3; ++d)
    oV[((size_t)n * FF + t) * 3 + d] = V[((size_t)n * 3 + d) * FF + t];
}

// ---------------------------------------------------------------------------
extern "C" void kernel_launch(void* const* d_in, const int* in_sizes, int n_in,
                              void* d_out, int out_size, void* d_ws, size_t ws_size,
                              hipStream_t stream) {
  const float* H      = (const float*)d_in[0];
  const float* cg_adj = (const float*)d_in[1];
  const float* cg_xyz = (const float*)d_in[2];
  const float* msg_w1 = (const float*)d_in[3];
  const float* msg_b1 = (const float*)d_in[4];
  const float* msg_w2 = (const float*)d_in[5];
  const float* msg_b2 = (const float*)d_in[6];
  const float* msg_wd = (const float*)d_in[7];
  const float* msg_bd = (const float*)d_in[8];
  const float* upd_wu = (const float*)d_in[9];
  const float* upd_wv = (const float*)d_in[10];
  const float* upd_w1 = (const float*)d_in[11];
  const float* upd_b1 = (const float*)d_in[12];
  const float* upd_w2 = (const float*)d_in[13];
  const float* upd_b2 = (const float*)d_in[14];
  float* out = (float*)d_out;

  // ---- workspace carve (floats) ----
  float* w = (float*)d_ws;
  float* Hs     = w; w += BN * FF;
  float* Vb0    = w; w += BN * 3 * FF;
  float* Vb1    = w; w += BN * 3 * FF;
  float* phi    = w; w += BN * F4;
  float* T1     = w; w += BN * FF;
  float* uv     = w; w += BN * 3 * FF;
  float* vv     = w; w += BN * 3 * FF;
  float* sstack = w; w += BN * F2;
  float* amat   = w; w += BN * F3;
  float* dinv   = w; w += BN;
  float* unitv  = w; w += NPAIR * 3;
  float* rbfe   = w; w += NPAIR * NRBF;
  float* ewv    = w; w += NPAIR;
  (void)ws_size; (void)in_sizes; (void)n_in; (void)out_size;

  auto launch_gemm = [&](const float* A, const float* W_, const float* bias, float* C,
                         int M, int K, int Nc, int act) {
    int tiles2 = (M / 16) * (Nc / 32);     // 16x32 per wave
    int blocks = (tiles2 + 3) / 4;         // 4 waves per 128-thread block
    gemm_wmma<<<blocks, 128, 0, stream>>>(A, W_, bias, C, M, K, Nc, tiles2, act);
  };

  // ---- one-time precompute ----
  copy_f32<<<(BN * FF + 127) / 128, 128, 0, stream>>>(H, Hs, BN * FF);
  zero_f32<<<(BN * 3 * FF + 127) / 128, 128, 0, stream>>>(Vb0, BN * 3 * FF);
  deg_kernel<<<(BN + 127) / 128, 128, 0, stream>>>(cg_adj, dinv);
  edge_precompute<<<(NPAIR + 127) / 128, 128, 0, stream>>>(cg_xyz, dinv, unitv, rbfe, ewv);

  float* Vin = Vb0;
  float* Vout = Vb1;

  for (int c = 0; c < 3; ++c) {
    const float* c_mw1 = msg_w1 + (size_t)c * FF * FF;
    const float* c_mb1 = msg_b1 + (size_t)c * FF;
    const float* c_mw2 = msg_w2 + (size_t)c * FF * F4;
    const float* c_mb2 = msg_b2 + (size_t)c * F4;
    const float* c_mwd = msg_wd + (size_t)c * NRBF * F4;
    const float* c_mbd = msg_bd + (size_t)c * F4;
    const float* c_uwu = upd_wu + (size_t)c * FF * FF;
    const float* c_uwv = upd_wv + (size_t)c * FF * FF;
    const float* c_uw1 = upd_w1 + (size_t)c * F2 * FF;
    const float* c_ub1 = upd_b1 + (size_t)c * FF;
    const float* c_uw2 = upd_w2 + (size_t)c * FF * F3;
    const float* c_ub2 = upd_b2 + (size_t)c * F3;

    // message MLP: phi = swish(Hs @ w1 + b1) @ w2 + b2
    launch_gemm(Hs, c_mw1, c_mb1, T1, BN, FF, FF, /*swish*/1);
    launch_gemm(T1, c_mw2, c_mb2, phi, BN, FF, F4, 0);

    // edge messages + segment sums (per-node gather; no atomics)
    message_kernel<<<BN, 128, 0, stream>>>(phi, Hs, Vin, Vout,
                                           unitv, rbfe, ewv, c_mwd, c_mbd);

    // update block
    launch_gemm(Vout, c_uwu, nullptr, uv, BN * 3, FF, FF, 0);  // u_v
    launch_gemm(Vout, c_uwv, nullptr, vv, BN * 3, FF, FF, 0);  // v_v
    vnorm_stack<<<BN, 128, 0, stream>>>(Hs, vv, sstack);
    launch_gemm(sstack, c_uw1, c_ub1, T1, BN, F2, FF, /*swish*/1);
    launch_gemm(T1, c_uw2, c_ub2, amat, BN, FF, F3, 0);
    final_update<<<BN, 128, 0, stream>>>(Hs, Vout, uv, vv, amat);

    float* tmp = Vin; Vin = Vout; Vout = tmp;
  }

  // Vin now holds the final V
  write_out<<<BN, 128, 0, stream>>>(Hs, Vin, out);
}